// GraphDecoder_34548716929568
// MI455X (gfx1250) — compile-verified
//
#include <hip/hip_runtime.h>
#include <hip/hip_bf16.h>
#include <math.h>
#include <stdint.h>

typedef __attribute__((ext_vector_type(2))) float v2f;
typedef __attribute__((ext_vector_type(8))) float v8f;

#define NTOT 8192
#define DDIM 256
#define TM 128
#define TN 64
#define KC 32
#define LDSTRIDE 36   // padded row stride (floats): 16B-aligned rows, conflict-free frag loads

// ---------------- Kernel 1: per-row inverse norms (eps-clamped) ----------------
__global__ __launch_bounds__(256) void row_inv_norms(const float* __restrict__ Z,
                                                     float* __restrict__ inv_norm) {
    int row  = blockIdx.x * 8 + (threadIdx.x >> 5);   // 8 waves per block, 1 row per wave
    int lane = threadIdx.x & 31;
    const float4* zr = (const float4*)(Z + (size_t)row * DDIM);
    float s = 0.0f;
    #pragma unroll
    for (int i = 0; i < 2; ++i) {
        float4 v = zr[lane + 32 * i];                 // 64 float4 per row, 2 per lane
        s += v.x * v.x + v.y * v.y + v.z * v.z + v.w * v.w;
    }
    #pragma unroll
    for (int off = 16; off > 0; off >>= 1) s += __shfl_xor(s, off, 32);
    if (lane == 0) {
        float n = fmaxf(sqrtf(s), 1e-8f);
        inv_norm[row] = 1.0f / n;
    }
}

// direct global->LDS async copy, 16B per lane, tracked by ASYNCcnt
__device__ __forceinline__ void async_copy_b128(const float* gptr, const float* lptr) {
    uint32_t lds = (uint32_t)(uintptr_t)lptr;        // low 32 bits of generic shared ptr = LDS offset
    uint64_t ga  = (uint64_t)(uintptr_t)gptr;
    asm volatile("global_load_async_to_lds_b128 %0, %1, off"
                 :: "v"(lds), "v"(ga) : "memory");
}

// ---------------- Kernel 2: G = Z Z^T via f32 WMMA, async-LDS double buffered ----------------
__global__ __launch_bounds__(256) void gram_sigmoid_cosine(const float* __restrict__ Z,
                                                           const int* __restrict__ batch,
                                                           const float* __restrict__ inv_norm,
                                                           float* __restrict__ A_out,
                                                           float* __restrict__ S_out) {
    __shared__ float As[2][TM * LDSTRIDE];
    __shared__ float Bs[2][TN * LDSTRIDE];

    const int m0   = blockIdx.y * TM;
    const int n0   = blockIdx.x * TN;
    const int tid  = threadIdx.x;
    const int wave = tid >> 5;
    const int lane = tid & 31;
    const int l16  = lane & 15;
    const int hi   = lane >> 4;        // 0: K pair {0,1}, 1: K pair {2,3} (ISA 16x4 layout)

    v8f c0 = {}, c1 = {}, c2 = {}, c3 = {};

    const int ldr = tid >> 3;          // loader row 0..31
    const int ldc = (tid & 7) * 4;     // loader float4 column

    const float* gA0 = Z + (size_t)(m0 + ldr) * DDIM + ldc;
    const float* gB0 = Z + (size_t)(n0 + ldr) * DDIM + ldc;

    // Issue one chunk's staging: 6 async b128 instrs per wave (4 for A, 2 for B)
    auto issue_chunk = [&](int chunk, int buf) {
        const float* ga = gA0 + chunk * KC;
        const float* gb = gB0 + chunk * KC;
        #pragma unroll
        for (int rr = 0; rr < 4; ++rr)
            async_copy_b128(ga + (size_t)(32 * rr) * DDIM,
                            &As[buf][(ldr + 32 * rr) * LDSTRIDE + ldc]);
        #pragma unroll
        for (int rr = 0; rr < 2; ++rr)
            async_copy_b128(gb + (size_t)(32 * rr) * DDIM,
                            &Bs[buf][(ldr + 32 * rr) * LDSTRIDE + ldc]);
    };

    issue_chunk(0, 0);

    #pragma unroll
    for (int i = 0; i < DDIM / KC; ++i) {          // 8 chunks
        const int cur = i & 1;
        if (i < DDIM / KC - 1) {
            issue_chunk(i + 1, (i + 1) & 1);       // prefetch next chunk into other buffer
            // in-order completion: <=6 outstanding means chunk i has fully landed
            asm volatile("s_wait_asynccnt 0x6" ::: "memory");
        } else {
            asm volatile("s_wait_asynccnt 0x0" ::: "memory");
        }
        __syncthreads();                           // publish LDS across waves

        const float* ap = &As[cur][(wave * 16 + l16) * LDSTRIDE + 2 * hi];
        const float* bp = &Bs[cur][l16 * LDSTRIDE + 2 * hi];

        #pragma unroll
        for (int kk = 0; kk < KC; kk += 4) {
            float2 af = *(const float2*)(ap + kk);
            v2f a; a.x = af.x; a.y = af.y;

            float2 b0f = *(const float2*)(bp + kk);
            float2 b1f = *(const float2*)(bp + 16 * LDSTRIDE + kk);
            float2 b2f = *(const float2*)(bp + 32 * LDSTRIDE + kk);
            float2 b3f = *(const float2*)(bp + 48 * LDSTRIDE + kk);
            v2f b0; b0.x = b0f.x; b0.y = b0f.y;
            v2f b1; b1.x = b1f.x; b1.y = b1f.y;
            v2f b2; b2.x = b2f.x; b2.y = b2f.y;
            v2f b3; b3.x = b3f.x; b3.y = b3f.y;

            c0 = __builtin_amdgcn_wmma_f32_16x16x4_f32(false, a, false, b0, (short)0, c0, false, false);
            c1 = __builtin_amdgcn_wmma_f32_16x16x4_f32(false, a, false, b1, (short)0, c1, false, false);
            c2 = __builtin_amdgcn_wmma_f32_16x16x4_f32(false, a, false, b2, (short)0, c2, false, false);
            c3 = __builtin_amdgcn_wmma_f32_16x16x4_f32(false, a, false, b3, (short)0, c3, false, false);
        }
        if (i < DDIM / KC - 1)
            __syncthreads();                       // all waves done reading `cur` before it is overwritten
    }

    // ---- epilogue: A = sigmoid(G); S = same-graph ? G*inv_i*inv_j : 0 ----
    const int mbase = m0 + wave * 16 + hi * 8;     // accumulator VGPR v -> row mbase + v
    int   bm[8];
    float im[8];
    #pragma unroll
    for (int v = 0; v < 8; ++v) {
        bm[v] = batch[mbase + v];
        im[v] = inv_norm[mbase + v];
    }

    v8f acc[4] = {c0, c1, c2, c3};
    #pragma unroll
    for (int t = 0; t < 4; ++t) {
        const int   n   = n0 + t * 16 + l16;
        const int   bn  = batch[n];
        const float in_ = inv_norm[n];
        #pragma unroll
        for (int v = 0; v < 8; ++v) {
            const size_t idx = (size_t)(mbase + v) * NTOT + n;
            const float g = acc[t][v];
            A_out[idx] = 1.0f / (1.0f + expf(-g));
            S_out[idx] = (bm[v] == bn) ? g * im[v] * in_ : 0.0f;
        }
    }
}

extern "C" void kernel_launch(void* const* d_in, const int* in_sizes, int n_in,
                              void* d_out, int out_size, void* d_ws, size_t ws_size,
                              hipStream_t stream) {
    const float* Z     = (const float*)d_in[0];
    const int*   batch = (const int*)d_in[1];
    float* A_out = (float*)d_out;                                // [N, N]
    float* S_out = (float*)d_out + (size_t)NTOT * NTOT;          // [N, N]
    float* inv_norm = (float*)d_ws;                              // [N]

    row_inv_norms<<<NTOT / 8, 256, 0, stream>>>(Z, inv_norm);

    dim3 grid(NTOT / TN, NTOT / TM);   // (128, 64)
    gram_sigmoid_cosine<<<grid, 256, 0, stream>>>(Z, batch, inv_norm, A_out, S_out);
}